// EdgeSAGEConv_81372450390264
// MI455X (gfx1250) — compile-verified
//
#include <hip/hip_runtime.h>
#include <hip/hip_bf16.h>

#define N_NODES 100000
#define N_EDGES 1600000
#define D_IN    256
#define D_EDGE  16
#define D_OUT   128

typedef __attribute__((ext_vector_type(2))) float v2f;
typedef __attribute__((ext_vector_type(8))) float v8f;

// ---------------------------------------------------------------------------
// Kernel 1: zero per-dst edge counts
// ---------------------------------------------------------------------------
__global__ void zero_cnt_kernel(int* __restrict__ cnt, int n) {
    int i = blockIdx.x * blockDim.x + threadIdx.x;
    if (i < n) cnt[i] = 0;
}

// ---------------------------------------------------------------------------
// Kernel 2: histogram of dst -> cnt
// ---------------------------------------------------------------------------
__global__ void count_edges_kernel(const int* __restrict__ dst,
                                   int* __restrict__ cnt, int e_total) {
    int e = blockIdx.x * blockDim.x + threadIdx.x;
    if (e < e_total) atomicAdd(&cnt[dst[e]], 1);
}

// ---------------------------------------------------------------------------
// Kernel 3: fused dual GEMM with fp32 WMMA (16x16x4).
//   out_self [N,128]  = X @ W_self + b_self   (written to d_out)
//   out_neigh[N,128]  = X @ W_neigh           (written to ws, h_src)
// Block: 512 threads = 16 waves. Waves 0..7 -> W_self tiles, 8..15 -> W_neigh.
// Each wave computes one 16x16 output tile; X tile (16x256) staged in LDS.
// ---------------------------------------------------------------------------
#define XS_STRIDE 257   // 256 + 1 pad: removes LDS bank conflicts on column reads

__global__ void __launch_bounds__(512)
gemm_dual_wmma_kernel(const float* __restrict__ X,
                      const float* __restrict__ W_self,
                      const float* __restrict__ W_neigh,
                      const float* __restrict__ b_self,
                      float* __restrict__ out_self,
                      float* __restrict__ out_neigh) {
    __shared__ float xs[16 * XS_STRIDE];

    const int rowBase = blockIdx.x * 16;

    // Cooperative load of the 16x256 X tile (coalesced, 8 floats/thread).
    for (int idx = threadIdx.x; idx < 16 * D_IN; idx += 512) {
        const int r = idx >> 8;        // / 256
        const int c = idx & 255;       // % 256
        xs[r * XS_STRIDE + c] = X[(size_t)(rowBase + r) * D_IN + c];
    }
    __syncthreads();

    const int wave  = threadIdx.x >> 5;      // 0..15
    const int lane  = threadIdx.x & 31;
    const int ntile = wave & 7;              // which 16-col tile of D_OUT
    const bool isSelf = (wave < 8);

    const float* __restrict__ W = isSelf ? W_self : W_neigh;

    const int m     = lane & 15;             // A row / D column-lane id
    const int khalf = lane >> 4;             // 0: K+0..1, 1: K+2..3
    const int col   = ntile * 16 + m;        // output column (B's N)

    v8f acc = {};
    const float* xrow = &xs[m * XS_STRIDE];

    #pragma unroll 4
    for (int kb = 0; kb < D_IN; kb += 4) {
        const int k0 = kb + 2 * khalf;
        v2f a, b;
        a.x = xrow[k0 + 0];
        a.y = xrow[k0 + 1];
        b.x = W[(size_t)(k0 + 0) * D_OUT + col];
        b.y = W[(size_t)(k0 + 1) * D_OUT + col];
        // D = A(16x4,f32) x B(4x16,f32) + C  -> v_wmma_f32_16x16x4_f32
        acc = __builtin_amdgcn_wmma_f32_16x16x4_f32(
            /*neg_a=*/false, a, /*neg_b=*/false, b,
            /*c_mod=*/(short)0, acc, /*reuse_a=*/false, /*reuse_b=*/false);
    }

    const float bias = isSelf ? b_self[col] : 0.0f;
    float* __restrict__ outp = isSelf ? out_self : out_neigh;

    // C/D layout: VGPR r holds M=r (lanes 0-15) / M=r+8 (lanes 16-31), N=lane&15
    #pragma unroll
    for (int r = 0; r < 8; ++r) {
        const int row = rowBase + r + 8 * khalf;
        outp[(size_t)row * D_OUT + col] = acc[r] + bias;
    }
}

// ---------------------------------------------------------------------------
// Kernel 4: per-edge scatter with fused edge projection + segment-mean.
// One wave per edge. W_edge (16x128 = 8KB) lives in LDS.
//   contribution[d, c] += (h_src[src, c] + sum_k xe[k]*W_edge[k, c]) / cnt[d]
// ---------------------------------------------------------------------------
#define EDGES_PER_BLOCK 8   // 8 waves x 32 lanes = 256 threads

__global__ void __launch_bounds__(256)
edge_scatter_kernel(const float* __restrict__ X_edge,
                    const float* __restrict__ h_src,
                    const int* __restrict__ src,
                    const int* __restrict__ dst,
                    const int* __restrict__ cnt,
                    const float* __restrict__ W_edge,
                    float* __restrict__ out, int e_total) {
    __shared__ float wl[D_EDGE * D_OUT];  // 8 KB
    for (int i = threadIdx.x; i < D_EDGE * D_OUT; i += 256) wl[i] = W_edge[i];
    __syncthreads();

    const int wave = threadIdx.x >> 5;
    const int lane = threadIdx.x & 31;
    const int e = blockIdx.x * EDGES_PER_BLOCK + wave;
    if (e >= e_total) return;

    const int s = src[e];
    const int d = dst[e];
    const float scale = 1.0f / (float)cnt[d];   // cnt >= 1: this edge exists

    // Broadcast the 16 edge features across the wave via shuffles.
    const float xv = X_edge[(size_t)e * D_EDGE + (lane & 15)];
    float xe[D_EDGE];
    #pragma unroll
    for (int k = 0; k < D_EDGE; ++k) xe[k] = __shfl(xv, k, 32);

    const float* __restrict__ hs = h_src + (size_t)s * D_OUT;
    float* __restrict__ od = out + (size_t)d * D_OUT;

    #pragma unroll
    for (int j = 0; j < 4; ++j) {
        const int c = lane + j * 32;
        float v = hs[c];
        #pragma unroll
        for (int k = 0; k < D_EDGE; ++k) v = fmaf(xe[k], wl[k * D_OUT + c], v);
        atomicAdd(&od[c], v * scale);   // global_atomic_add_f32
    }
}

// ---------------------------------------------------------------------------
// Launch
// ---------------------------------------------------------------------------
extern "C" void kernel_launch(void* const* d_in, const int* in_sizes, int n_in,
                              void* d_out, int out_size, void* d_ws, size_t ws_size,
                              hipStream_t stream) {
    const float* X       = (const float*)d_in[0];
    const float* X_edge  = (const float*)d_in[1];
    const float* W_neigh = (const float*)d_in[2];
    const float* W_self  = (const float*)d_in[3];
    const float* b_self  = (const float*)d_in[4];
    const float* W_edge  = (const float*)d_in[5];
    const int*   src     = (const int*)d_in[6];
    const int*   dst     = (const int*)d_in[7];
    float* out = (float*)d_out;

    // Workspace layout: h_src [N_NODES * D_OUT] f32, then cnt [N_NODES] int
    float* h_src = (float*)d_ws;
    int*   cnt   = (int*)((char*)d_ws + (size_t)N_NODES * D_OUT * sizeof(float));

    zero_cnt_kernel<<<(N_NODES + 255) / 256, 256, 0, stream>>>(cnt, N_NODES);
    count_edges_kernel<<<(N_EDGES + 255) / 256, 256, 0, stream>>>(dst, cnt, N_EDGES);

    gemm_dual_wmma_kernel<<<N_NODES / 16, 512, 0, stream>>>(
        X, W_self, W_neigh, b_self, out, h_src);

    edge_scatter_kernel<<<(N_EDGES + EDGES_PER_BLOCK - 1) / EDGES_PER_BLOCK, 256, 0, stream>>>(
        X_edge, h_src, src, dst, cnt, W_edge, out, N_EDGES);
}